// LSMMACEMetaEncoder_16819091931678
// MI455X (gfx1250) — compile-verified
//
#include <hip/hip_runtime.h>
#include <hip/hip_bf16.h>
#include <math.h>

// ---------------------------------------------------------------------------
// MACE-style GNN forward for MI455X (gfx1250, wave32, WMMA + TDM).
// Dense GEMMs: V_WMMA_F32_16X16X4_F32, B panels DMA'd to LDS by the Tensor
// Data Mover (tensor_load_to_lds) with double buffering (TENSORcnt pipeline).
// Edge work chunked so the [ECH,896] radial-weight tile stays L2-resident.
// ---------------------------------------------------------------------------

#define N_NODES 10000
#define NP      10112          // nodes padded to multiple of 128 (block M-tile)
#define N_EDGES 160000
#define ECH     16000          // edge chunk (mult of 128); w-chunk = 57 MB < L2
#define HS      256
#define HV      128
#define RH      64
#define NB      64
#define DOUT    (2*HS + 3*HV)  // 896
#define CS_IN   128
#define CV_IN   32
#define NL      6
#define RCUT    5.0f

typedef __attribute__((ext_vector_type(2))) float v2f;
typedef __attribute__((ext_vector_type(8))) float v8f;
typedef __attribute__((ext_vector_type(4))) unsigned int u32x4;
typedef __attribute__((ext_vector_type(4))) int i32x4;
typedef __attribute__((ext_vector_type(8))) int i32x8;

#if __has_builtin(__builtin_amdgcn_tensor_load_to_lds)
#define HAVE_TDM 1
#else
#define HAVE_TDM 0
#endif

#if HAVE_TDM
// ---------------------------------------------------------------------------
// Issue one TDM descriptor: load a [rows x width] fp32 tile (row stride = ldn
// elements) from global into LDS at byte offset lds_off. D# layout per CDNA5
// ISA ch.8: group0 = {count, lds_addr, global_addr, type=2}; group1 carries
// data_size / tensor dims / tile dims / strides. Groups 2,3 zero (2-D tensor).
// ---------------------------------------------------------------------------
__device__ __forceinline__ void tdm_load_panel(const float* gsrc, unsigned lds_off,
                                               int rows, int width, int ldn)
{
    unsigned long long ga = (unsigned long long)(uintptr_t)gsrc;
    u32x4 g0 = {};
    g0[0] = 1u;                                                  // count=1, user D#
    g0[1] = lds_off;                                             // lds_addr (bytes)
    g0[2] = (unsigned)ga;                                        // global_addr[31:0]
    g0[3] = (unsigned)((ga >> 32) & 0x1FFFFFFu) | (2u << 30);    // addr[56:32] | type=2
    i32x8 g1 = {};
    g1[0] = (int)(2u << 16);                                     // wg_mask=0, data_size=4B
    g1[1] = (int)(((unsigned)ldn & 0xFFFFu) << 16);              // tensor_dim0[15:0]
    g1[2] = (int)(((unsigned)ldn >> 16) |
                  (((unsigned)rows & 0xFFFFu) << 16));           // dim0 hi | tensor_dim1 lo
    g1[3] = (int)((((unsigned)rows) >> 16) |
                  (((unsigned)width & 0xFFFFu) << 16));          // dim1 hi | tile_dim0
    g1[4] = (int)((unsigned)rows & 0xFFFFu);                     // tile_dim1 (tile_dim2=0)
    g1[5] = (int)(unsigned)ldn;                                  // tensor_dim0_stride[31:0]
    g1[6] = 0;
    g1[7] = 0;
    i32x4 z4 = {};
#if defined(__clang_major__) && __clang_major__ >= 23
    i32x8 z8 = {};
    __builtin_amdgcn_tensor_load_to_lds(g0, g1, z4, z4, z8, 0);
#else
    __builtin_amdgcn_tensor_load_to_lds(g0, g1, z4, z4, 0);
#endif
}
#endif

// One K=4 WMMA micro-step: A pair from global (immediate offset), 4 B pairs
// from the LDS panel (bank-conflict-free, immediate offsets).
#define GEMM_STEP(APTR, BPTR, KK)                                                 \
    {                                                                             \
        v2f a;                                                                    \
        a.x = (APTR)[(KK)];                                                       \
        a.y = (APTR)[(KK) + 1];                                                   \
        const float* bq = (BPTR) + ((KK) + khalf) * 64 + mrow;                    \
        v2f b0, b1, b2, b3;                                                       \
        b0.x = bq[0];  b0.y = bq[64];                                             \
        b1.x = bq[16]; b1.y = bq[80];                                             \
        b2.x = bq[32]; b2.y = bq[96];                                             \
        b3.x = bq[48]; b3.y = bq[112];                                            \
        acc0 = __builtin_amdgcn_wmma_f32_16x16x4_f32(false, a, false, b0, (short)0, acc0, false, false); \
        acc1 = __builtin_amdgcn_wmma_f32_16x16x4_f32(false, a, false, b1, (short)0, acc1, false, false); \
        acc2 = __builtin_amdgcn_wmma_f32_16x16x4_f32(false, a, false, b2, (short)0, acc2, false, false); \
        acc3 = __builtin_amdgcn_wmma_f32_16x16x4_f32(false, a, false, b3, (short)0, acc3, false, false); \
    }

// ---------------------------------------------------------------------------
// fp32 WMMA GEMM:  D[M,N] = epilogue(A[M,K] @ B[K,N]),  A lda=K, B ldb=N.
//   Block = 256 thr = 8 waves stacked along M (128 rows) sharing one 64-wide
//   B panel. B staged to LDS in 64-row K-chunks by TDM, double-buffered.
//   EPI: 0 = store, 1 = SiLU, 2 = D = acc + Cadd.
// Requires: M % 128 == 0, N % 64 == 0, K % 4 == 0.
// ---------------------------------------------------------------------------
template <int EPI>
__global__ __launch_bounds__(256)
void wmma_gemm_f32(const float* __restrict__ A, const float* __restrict__ B,
                   const float* __restrict__ Cadd, float* __restrict__ D,
                   int M, int N, int K)
{
    __shared__ float bp[2][64 * 64];                // 2 x 16 KB double buffer
    const int lane  = threadIdx.x & 31;
    const int wave  = threadIdx.x >> 5;
    const int tn    = blockIdx.x;                   // 64-col panel index
    const int tm    = blockIdx.y * 8 + wave;        // 16-row tile index
    const int mrow  = lane & 15;
    const int khalf = (lane >> 4) << 1;             // lanes 0-15: K+0,1 ; 16-31: K+2,3
    const int nchunks = (K + 63) >> 6;
    const float* Bsrc = B + (size_t)tn * 64;

#if HAVE_TDM
    if (wave == 0) {                                // prologue DMA: chunk 0
        int kc0 = K < 64 ? K : 64;
        tdm_load_panel(Bsrc, (unsigned)(uintptr_t)(const void*)&bp[0][0], kc0, 64, N);
    }
#endif

    v8f acc0 = {}, acc1 = {}, acc2 = {}, acc3 = {};
    const float* Ap = A + (size_t)(tm * 16 + mrow) * K + khalf;

    for (int c = 0; c < nchunks; ++c) {
        const int k0 = c << 6;
        const int kc = (K - k0) < 64 ? (K - k0) : 64;
#if HAVE_TDM
        if (wave == 0) __builtin_amdgcn_s_wait_tensorcnt(0);   // chunk c landed
        __syncthreads();                                        // publish LDS
        if (wave == 0 && (c + 1) < nchunks) {                   // overlap next DMA
            int kn = (K - (k0 + 64)) < 64 ? (K - (k0 + 64)) : 64;
            tdm_load_panel(Bsrc + (size_t)(k0 + 64) * N,
                           (unsigned)(uintptr_t)(const void*)&bp[(c + 1) & 1][0],
                           kn, 64, N);
        }
#else
        __syncthreads();
        for (int i = threadIdx.x; i < kc * 64; i += 256)        // cooperative fallback
            bp[c & 1][i] = Bsrc[(size_t)(k0 + (i >> 6)) * N + (i & 63)];
        __syncthreads();
#endif
        const float* bpc = bp[c & 1];
        const float* Apc = Ap;
        if (kc == 64) {
            // fast path: compile-time trip count -> full unroll, immediate
            // load offsets, loads pipelined across 64 WMMAs per chunk.
#pragma unroll
            for (int kk = 0; kk < 64; kk += 4) {
                GEMM_STEP(Apc, bpc, kk)
            }
        } else {
            for (int kk = 0; kk < kc; kk += 4) {
                GEMM_STEP(Apc, bpc, 0)
                Apc += 4;
                bpc += 4 * 64;
            }
        }
        Ap += kc;
        __syncthreads();                                        // reads done before reuse
    }

    // C/D layout: VGPR j -> M = tm*16 + j + (lane>=16 ? 8 : 0), N = tn*64 + mrow + t*16
    const int row0 = tm * 16 + ((lane >> 4) << 3);
    const int colb = tn * 64 + mrow;
#pragma unroll
    for (int t = 0; t < 4; ++t) {
        v8f acc = (t == 0) ? acc0 : (t == 1) ? acc1 : (t == 2) ? acc2 : acc3;
        const int col = colb + t * 16;
#pragma unroll
        for (int j = 0; j < 8; ++j) {
            float val = acc[j];
            if (EPI == 1) val = val / (1.0f + __expf(-val));             // SiLU
            if (EPI == 2) val += Cadd[(size_t)(row0 + j) * N + col];     // residual
            D[(size_t)(row0 + j) * N + col] = val;
        }
    }
}

// ---------------------------------------------------------------------------
// Per-edge geometry: r, Y1 = sqrt(3)*dhat, T2 = dhat dhat^T - I/3,
// edge_attr = gaussian_basis(r) * envelope(r).  One thread per edge.
// ---------------------------------------------------------------------------
__global__ void edge_geom_kernel(const float* __restrict__ pos,
                                 const float* __restrict__ shifts,
                                 const int* __restrict__ ei,
                                 float* __restrict__ Y1, float* __restrict__ T2,
                                 float* __restrict__ eattr)
{
    int e = blockIdx.x * blockDim.x + threadIdx.x;
    if (e >= N_EDGES) return;
    int s = ei[e];
    int d = ei[N_EDGES + e];
    float ex = pos[d * 3 + 0] - pos[s * 3 + 0] + shifts[e * 3 + 0];
    float ey = pos[d * 3 + 1] - pos[s * 3 + 1] + shifts[e * 3 + 1];
    float ez = pos[d * 3 + 2] - pos[s * 3 + 2] + shifts[e * 3 + 2];
    float r = sqrtf(ex * ex + ey * ey + ez * ez + 1e-12f);
    float inv = 1.0f / fmaxf(r, 1e-8f);
    float dx = ex * inv, dy = ey * inv, dz = ez * inv;

    const float S3 = 1.7320508075688772f;
    Y1[e * 3 + 0] = S3 * dx;
    Y1[e * 3 + 1] = S3 * dy;
    Y1[e * 3 + 2] = S3 * dz;

    const float third = 1.0f / 3.0f;
    float* t = T2 + (size_t)e * 9;
    t[0] = dx * dx - third; t[1] = dx * dy;         t[2] = dx * dz;
    t[3] = dy * dx;         t[4] = dy * dy - third; t[5] = dy * dz;
    t[6] = dz * dx;         t[7] = dz * dy;         t[8] = dz * dz - third;

    float u2 = (r / RCUT) * (r / RCUT);
    u2 = fminf(fmaxf(u2, 0.0f), 1.0f - 1e-6f);
    float env = (r < RCUT) ? __expf(1.0f - 1.0f / (1.0f - u2)) : 0.0f;

    const float width = RCUT / (float)NB;
    const float cstep = RCUT / (float)(NB - 1);
    float* ea = eattr + (size_t)e * NB;
#pragma unroll 8
    for (int b = 0; b < NB; ++b) {
        float z = (r - cstep * (float)b) / width;
        ea[b] = __expf(-0.5f * z * z) * env;
    }
}

// ---------------------------------------------------------------------------
// Scalar messages: m_s[e,h] = w0*ss[src,h] + w1*(vs[src,h,:].Y1[e]); scatter.
// ---------------------------------------------------------------------------
__global__ void msg_s_kernel(const float* __restrict__ w,
                             const float* __restrict__ ss,
                             const float* __restrict__ vs0, const float* __restrict__ vs1,
                             const float* __restrict__ vs2,
                             const float* __restrict__ Y1,
                             const int* __restrict__ src, const int* __restrict__ dst,
                             float* __restrict__ ns, int e0, int ne)
{
    int idx = blockIdx.x * blockDim.x + threadIdx.x;
    if (idx >= ne * HS) return;
    int el = idx >> 8;
    int h  = idx & (HS - 1);
    int e  = e0 + el;
    int sn = src[e], dn = dst[e];
    const float* we = w + (size_t)el * DOUT;
    float w0 = we[h];
    float w1 = we[HS + h];
    float y0 = Y1[e * 3 + 0], y1 = Y1[e * 3 + 1], y2 = Y1[e * 3 + 2];
    size_t sh = (size_t)sn * HS + h;
    float vdot = vs0[sh] * y0 + vs1[sh] * y1 + vs2[sh] * y2;
    float m = w0 * ss[sh] + w1 * vdot;
    atomicAdd(&ns[(size_t)dn * HS + h], m);
}

// ---------------------------------------------------------------------------
// Vector messages:
//  m_v[e,h,d] = w2*sv[src,h]*Y1[e,d] + w3*vv[src,h,d] + w4*(T2[e].vv[src,h,:])_d
// ---------------------------------------------------------------------------
__global__ void msg_v_kernel(const float* __restrict__ w,
                             const float* __restrict__ sv,
                             const float* __restrict__ vv0, const float* __restrict__ vv1,
                             const float* __restrict__ vv2,
                             const float* __restrict__ Y1, const float* __restrict__ T2,
                             const int* __restrict__ src, const int* __restrict__ dst,
                             float* __restrict__ nv0, float* __restrict__ nv1,
                             float* __restrict__ nv2, int e0, int ne)
{
    int idx = blockIdx.x * blockDim.x + threadIdx.x;
    if (idx >= ne * HV) return;
    int el = idx >> 7;
    int h  = idx & (HV - 1);
    int e  = e0 + el;
    int sn = src[e], dn = dst[e];
    const float* we = w + (size_t)el * DOUT + 2 * HS;
    float w2 = we[h];
    float w3 = we[HV + h];
    float w4 = we[2 * HV + h];
    float y0 = Y1[e * 3 + 0], y1 = Y1[e * 3 + 1], y2 = Y1[e * 3 + 2];
    const float* t = T2 + (size_t)e * 9;
    size_t sh = (size_t)sn * HV + h;
    float svh = sv[sh];
    float a0 = vv0[sh], a1 = vv1[sh], a2 = vv2[sh];
    float c0 = t[0] * a0 + t[1] * a1 + t[2] * a2;
    float c1 = t[3] * a0 + t[4] * a1 + t[5] * a2;
    float c2 = t[6] * a0 + t[7] * a1 + t[8] * a2;
    size_t dh = (size_t)dn * HV + h;
    atomicAdd(&nv0[dh], w2 * svh * y0 + w3 * a0 + w4 * c0);
    atomicAdd(&nv1[dh], w2 * svh * y1 + w3 * a1 + w4 * c1);
    atomicAdd(&nv2[dh], w2 * svh * y2 + w3 * a2 + w4 * c2);
}

// ---------------------------------------------------------------------------
// Input staging: s -> [N,128] contiguous; v [N,32,3] -> 3 de-interleaved [N,32].
// ---------------------------------------------------------------------------
__global__ void pack_in_kernel(const float* __restrict__ sfeat,
                               const float* __restrict__ vfeat,
                               float* __restrict__ sA,
                               float* __restrict__ v0, float* __restrict__ v1,
                               float* __restrict__ v2)
{
    int idx = blockIdx.x * blockDim.x + threadIdx.x;
    const int NS = N_NODES * CS_IN;
    const int NV = N_NODES * CV_IN * 3;
    if (idx < NS) {
        sA[idx] = sfeat[idx];
    } else if (idx < NS + NV) {
        int rel = idx - NS;
        int n = rel / (CV_IN * 3);
        int k = rel - n * (CV_IN * 3);
        int c = k / 3;
        int d = k - c * 3;
        float val = vfeat[rel];
        float* vp = (d == 0) ? v0 : (d == 1) ? v1 : v2;
        vp[n * CV_IN + c] = val;
    }
}

// ---------------------------------------------------------------------------
// Output packing: d_out = [ x (N x 640) | s_out (N x 256) | v_out (N x 128 x 3) ]
// ---------------------------------------------------------------------------
__global__ void pack_out_kernel(const float* __restrict__ s,
                                const float* __restrict__ v0, const float* __restrict__ v1,
                                const float* __restrict__ v2,
                                float* __restrict__ out)
{
    const size_t NX = (size_t)N_NODES * 640;
    const size_t NS = (size_t)N_NODES * HS;
    const size_t NV = (size_t)N_NODES * HV * 3;
    size_t idx = (size_t)blockIdx.x * blockDim.x + threadIdx.x;
    if (idx >= NX + NS + NV) return;
    float val;
    if (idx < NX) {
        int n = (int)(idx / 640);
        int c = (int)(idx - (size_t)n * 640);
        if (c < HS) {
            val = s[(size_t)n * HS + c];
        } else {
            int k = c - HS;
            int h = k / 3, d = k - h * 3;
            const float* vp = (d == 0) ? v0 : (d == 1) ? v1 : v2;
            val = vp[(size_t)n * HV + h];
        }
    } else if (idx < NX + NS) {
        val = s[idx - NX];
    } else {
        size_t r = idx - NX - NS;
        int n = (int)(r / (HV * 3));
        int k = (int)(r - (size_t)n * (HV * 3));
        int h = k / 3, d = k - h * 3;
        const float* vp = (d == 0) ? v0 : (d == 1) ? v1 : v2;
        val = vp[(size_t)n * HV + h];
    }
    out[idx] = val;
}

__global__ void zero_kernel(float* __restrict__ p, size_t n)
{
    size_t i = (size_t)blockIdx.x * blockDim.x + threadIdx.x;
    if (i < n) p[i] = 0.0f;
}

// ---------------------------------------------------------------------------
// Host orchestration
// ---------------------------------------------------------------------------
extern "C" void kernel_launch(void* const* d_in, const int* in_sizes, int n_in,
                              void* d_out, int out_size, void* d_ws, size_t ws_size,
                              hipStream_t stream)
{
    (void)in_sizes; (void)n_in; (void)out_size; (void)ws_size;

    const float* pos    = (const float*)d_in[0];
    const float* shifts = (const float*)d_in[1];
    const float* sfeat  = (const float*)d_in[2];
    const float* vfeat  = (const float*)d_in[3];
    const int*   ei     = (const int*)d_in[4 + 8 * NL];   // index 52
    float*       out    = (float*)d_out;

    float* wsf = (float*)d_ws;
    size_t off = 0;
    auto alloc = [&](size_t n) { float* p = wsf + off; off += n; return p; };

    float* Y1     = alloc((size_t)N_EDGES * 3);
    float* T2     = alloc((size_t)N_EDGES * 9);
    float* eattr  = alloc((size_t)N_EDGES * NB);
    float* hidden = alloc((size_t)ECH * RH);
    float* wbuf   = alloc((size_t)ECH * DOUT);
    float* ss     = alloc((size_t)NP * HS);
    float* sv     = alloc((size_t)NP * HV);
    float* vs[3]  = { alloc((size_t)NP * HS), alloc((size_t)NP * HS), alloc((size_t)NP * HS) };
    float* vv[3]  = { alloc((size_t)NP * HV), alloc((size_t)NP * HV), alloc((size_t)NP * HV) };
    float* ns     = alloc((size_t)NP * HS);           // ns..nv2 contiguous
    float* nv[3]  = { alloc((size_t)NP * HV), alloc((size_t)NP * HV), alloc((size_t)NP * HV) };
    float* sA     = alloc((size_t)NP * HS);
    float* sB     = alloc((size_t)NP * HS);
    float* vA[3]  = { alloc((size_t)NP * HV), alloc((size_t)NP * HV), alloc((size_t)NP * HV) };
    float* vB[3]  = { alloc((size_t)NP * HV), alloc((size_t)NP * HV), alloc((size_t)NP * HV) };

    auto gemm = [&](const float* A, const float* B, const float* C, float* D,
                    int M, int N, int K, int epi) {
        dim3 grid(N >> 6, M >> 7);   // N/64 panels x M/128 row groups
        if (epi == 0)      wmma_gemm_f32<0><<<grid, 256, 0, stream>>>(A, B, C, D, M, N, K);
        else if (epi == 1) wmma_gemm_f32<1><<<grid, 256, 0, stream>>>(A, B, C, D, M, N, K);
        else               wmma_gemm_f32<2><<<grid, 256, 0, stream>>>(A, B, C, D, M, N, K);
    };

    {
        int tot = N_NODES * (CS_IN + CV_IN * 3);
        pack_in_kernel<<<(tot + 255) / 256, 256, 0, stream>>>(sfeat, vfeat, sA, vA[0], vA[1], vA[2]);
        edge_geom_kernel<<<(N_EDGES + 255) / 256, 256, 0, stream>>>(pos, shifts, ei, Y1, T2, eattr);
    }

    float* s_cur = sA;  float* s_nxt = sB;
    float* v_cur[3] = { vA[0], vA[1], vA[2] };
    float* v_nxt[3] = { vB[0], vB[1], vB[2] };

    const int* src = ei;
    const int* dst = ei + N_EDGES;

    for (int l = 0; l < NL; ++l) {
        const int base = 4 + 8 * l;
        const float* ws_s = (const float*)d_in[base + 0];
        const float* ws_v = (const float*)d_in[base + 1];
        const float* wv_s = (const float*)d_in[base + 2];
        const float* wv_v = (const float*)d_in[base + 3];
        const float* r1   = (const float*)d_in[base + 4];
        const float* r2   = (const float*)d_in[base + 5];
        const float* lo_s = (const float*)d_in[base + 6];
        const float* lo_v = (const float*)d_in[base + 7];
        const int Cs = l ? HS : CS_IN;
        const int Cv = l ? HV : CV_IN;

        gemm(s_cur, ws_s, nullptr, ss, NP, HS, Cs, 0);
        gemm(s_cur, ws_v, nullptr, sv, NP, HV, Cs, 0);
        for (int d = 0; d < 3; ++d) gemm(v_cur[d], wv_s, nullptr, vs[d], NP, HS, Cv, 0);
        for (int d = 0; d < 3; ++d) gemm(v_cur[d], wv_v, nullptr, vv[d], NP, HV, Cv, 0);

        {
            size_t n = (size_t)NP * (HS + 3 * HV);
            zero_kernel<<<(unsigned)((n + 255) / 256), 256, 0, stream>>>(ns, n);
        }

        for (int e0 = 0; e0 < N_EDGES; e0 += ECH) {
            gemm(eattr + (size_t)e0 * NB, r1, nullptr, hidden, ECH, RH, NB, 1);  // SiLU
            gemm(hidden, r2, nullptr, wbuf, ECH, DOUT, RH, 0);
            msg_s_kernel<<<(ECH * HS + 255) / 256, 256, 0, stream>>>(
                wbuf, ss, vs[0], vs[1], vs[2], Y1, src, dst, ns, e0, ECH);
            msg_v_kernel<<<(ECH * HV + 255) / 256, 256, 0, stream>>>(
                wbuf, sv, vv[0], vv[1], vv[2], Y1, T2, src, dst,
                nv[0], nv[1], nv[2], e0, ECH);
        }

        gemm(ns, lo_s, ss, s_nxt, NP, HS, HS, 2);
        for (int d = 0; d < 3; ++d) gemm(nv[d], lo_v, vv[d], v_nxt[d], NP, HV, HV, 2);

        float* ts = s_cur; s_cur = s_nxt; s_nxt = ts;
        for (int d = 0; d < 3; ++d) { float* tv = v_cur[d]; v_cur[d] = v_nxt[d]; v_nxt[d] = tv; }
    }

    {
        size_t tot = (size_t)N_NODES * (640 + HS + HV * 3);
        pack_out_kernel<<<(unsigned)((tot + 255) / 256), 256, 0, stream>>>(
            s_cur, v_cur[0], v_cur[1], v_cur[2], out);
    }
}